// ScaledDotProductAttention_47175920779376
// MI455X (gfx1250) — compile-verified
//
#include <hip/hip_runtime.h>

typedef __attribute__((ext_vector_type(2))) float v2f;
typedef __attribute__((ext_vector_type(8))) float v8f;

#define S_LEN    2048
#define D_DIM    128
#define MTILE    16
#define NTILES   (S_LEN / MTILE)     // 128 key tiles
#define ROWSTR   2050                // padded LDS row stride (floats), keeps 8B align
#define NWAVES   8

__device__ __forceinline__ v8f wmma4(v2f a, v2f b, v8f c) {
  // D = A(16x4 f32) * B(4x16 f32) + C(16x16 f32)
  return __builtin_amdgcn_wmma_f32_16x16x4_f32(
      /*neg_a=*/false, a, /*neg_b=*/false, b,
      /*c_mod=*/(short)0, c, /*reuse_a=*/false, /*reuse_b=*/false);
}

__global__ __launch_bounds__(256, 1)
void entmax15_attn_kernel(const float* __restrict__ Q,
                          const float* __restrict__ K,
                          const float* __restrict__ V,
                          float* __restrict__ O) {
  // 16 rows x 2048 scores (padded) -> 131,200 B of LDS (WGP has 320 KB)
  __shared__ float sS[MTILE * ROWSTR];

  const int tid   = threadIdx.x;
  const int wave  = tid >> 5;          // 0..7
  const int lane  = tid & 31;
  const int lrow  = lane & 15;         // N (or M for A-frag) within tile
  const int lhalf = lane >> 4;         // selects k-pair / row-half

  const int mt = blockIdx.x & (NTILES - 1);      // query tile 0..127
  const int bh = blockIdx.x >> 7;                // (b,h) flat, 0..15
  const size_t base = (size_t)bh * S_LEN * D_DIM;
  const int m0 = mt * MTILE;

  const float scale = 0.08838834764831845f;      // 1/sqrt(128)

  // ---------- preload Q A-fragments: 32 chunks of K=4 ----------
  // A layout (16x4 f32): lane m=lrow holds k = kbase + 2*lhalf, +1
  v2f afrag[32];
  {
    const float* qrow = Q + base + (size_t)(m0 + lrow) * D_DIM + 2 * lhalf;
#pragma unroll
    for (int kc = 0; kc < 32; ++kc)
      afrag[kc] = *(const v2f*)(qrow + kc * 4);
  }

  // ---------- phase 1: S = scale * Q K^T, 16 key-tiles per wave ----------
  for (int t = wave; t < NTILES; t += NWAVES) {
    const int n0 = t * MTILE;
    const float* krow = K + base + (size_t)(n0 + lrow) * D_DIM + 2 * lhalf;
    v8f acc = {};
#pragma unroll
    for (int kc = 0; kc < 32; ++kc) {
      v2f b = *(const v2f*)(krow + kc * 4);
      acc = wmma4(afrag[kc], b, acc);
    }
    // C layout: lane n=lrow, VGPR r -> m = r + 8*lhalf
#pragma unroll
    for (int r = 0; r < 8; ++r)
      sS[(r + 8 * lhalf) * ROWSTR + n0 + lrow] = acc[r] * scale;
  }

  // ---------- prefetch this wave's V column slice under the entmax phase ----
  {
    const float* vpf = V + base + wave * MTILE;  // 64B slice of each 512B row
    for (int kk = lane; kk < S_LEN; kk += 32)
      __builtin_prefetch(vpf + (size_t)kk * D_DIM, 0, 3);
  }
  __syncthreads();

  // ---------- phase 2: per-row entmax-1.5 (max, shift*0.5, Newton for tau) ----
  {
    // rows differ by 8 between half-waves -> LDS bank offset 8*2050 % 64 = 16
    // -> the two halves hit disjoint bank sets on every strided access.
    const int row = wave + 8 * lhalf;
    float* xr = &sS[row * ROWSTR];

    float mx = -3.402823466e38f;
#pragma unroll 4
    for (int j = 0; j < 128; ++j) mx = fmaxf(mx, xr[lrow + 16 * j]);
#pragma unroll
    for (int off = 1; off < 16; off <<= 1)
      mx = fmaxf(mx, __shfl_xor(mx, off, 32));   // reduce within 16-lane group

#pragma unroll 4
    for (int j = 0; j < 128; ++j) {
      const int c = lrow + 16 * j;
      xr[c] = (xr[c] - mx) * 0.5f;               // x <= 0, max(x) == 0
    }

    // Solve f(tau) = sum clip(x - tau)^2 = 1; f convex decreasing, tau* in (-1,0).
    // Guarded Newton: f' = -2 * sum clip(x - tau).
    float lo = -1.0f, hi = 0.0f;
    float tau = -0.5f;
    for (int it = 0; it < 12; ++it) {
      float fs = 0.0f, gs = 0.0f;
#pragma unroll 4
      for (int j = 0; j < 128; ++j) {
        float u = fmaxf(xr[lrow + 16 * j] - tau, 0.0f);
        fs = fmaf(u, u, fs);
        gs += u;
      }
#pragma unroll
      for (int off = 1; off < 16; off <<= 1) {
        fs += __shfl_xor(fs, off, 32);
        gs += __shfl_xor(gs, off, 32);
      }
      if (fs > 1.0f) lo = tau; else hi = tau;
      float tn = (gs > 1e-30f) ? tau + (fs - 1.0f) / (2.0f * gs)
                               : 0.5f * (lo + hi);
      tau = (tn > lo && tn < hi) ? tn : 0.5f * (lo + hi);
    }

#pragma unroll 4
    for (int j = 0; j < 128; ++j) {
      const int c = lrow + 16 * j;
      float u = fmaxf(xr[c] - tau, 0.0f);
      xr[c] = u * u;                             // attention weights in place
    }
  }
  __syncthreads();

  // ---------- phase 3: O = P @ V; wave w owns output cols [16w, 16w+16) ------
  {
    const int col = wave * MTILE + lrow;
    const float* vcol = V + base + col;          // V[k][col] = vcol[k * D]
    v8f acc = {};
    for (int kc = 0; kc < S_LEN / 4; ++kc) {
      const int kk = kc * 4 + 2 * lhalf;
      v2f a = *(const v2f*)&sS[lrow * ROWSTR + kk];   // P[m=lrow][kk, kk+1]
      v2f b;
      b.x = vcol[(size_t)kk * D_DIM];
      b.y = vcol[(size_t)(kk + 1) * D_DIM];
      acc = wmma4(a, b, acc);
    }
    float* orow = O + base + (size_t)m0 * D_DIM;
#pragma unroll
    for (int r = 0; r < 8; ++r)
      orow[(size_t)(r + 8 * lhalf) * D_DIM + col] = acc[r];
  }
}

extern "C" void kernel_launch(void* const* d_in, const int* in_sizes, int n_in,
                              void* d_out, int out_size, void* d_ws, size_t ws_size,
                              hipStream_t stream) {
  const float* q = (const float*)d_in[0];
  const float* k = (const float*)d_in[1];
  const float* v = (const float*)d_in[2];
  float* o = (float*)d_out;

  const int BH = in_sizes[0] / (S_LEN * D_DIM);  // B*H = 16
  dim3 grid(BH * NTILES);                        // 2048 workgroups
  entmax15_attn_kernel<<<grid, NWAVES * 32, 0, stream>>>(q, k, v, o);
}